// ChamferLoss2D_41824391529113
// MI455X (gfx1250) — compile-verified
//
#include <hip/hip_runtime.h>
#include <math.h>

typedef __attribute__((ext_vector_type(2))) float v2f;
typedef __attribute__((ext_vector_type(8))) float v8f;

#define NB 16
#define NP 4096
#define INF_BITS 0x7F800000
#define EPS 1e-12f

// ---------------------------------------------------------------------------
// Init: column-min buffer to +inf (as int bits; all distances are positive so
// signed-int ordering of the bit patterns == float ordering).
// ---------------------------------------------------------------------------
__global__ void cham_init(int* __restrict__ colmin) {
    int i = blockIdx.x * blockDim.x + threadIdx.x;
    if (i < NB * NP) colmin[i] = INF_BITS;
}

// ---------------------------------------------------------------------------
// Main: each workgroup = (batch n, 128-row slab). 8 waves; each wave owns a
// 16-row A tile (pre-scaled by -2) and streams all 256 column tiles of y,
// computing the 16x16 cross-term with V_WMMA_F32_16X16X4_F32 (K padded 2->4).
// Epilogue adds |x|^2+|y|^2, clamps at EPS, tracks row mins in registers and
// column mins via LDS int-min atomics, then merges LDS -> global colmin.
// ---------------------------------------------------------------------------
__global__ __launch_bounds__(256) void cham_main(const float* __restrict__ xs,
                                                 const float* __restrict__ ys,
                                                 int* __restrict__ colmin_g,
                                                 float* __restrict__ rowmin_g) {
    __shared__ int colminLDS[NP];

    const int n    = blockIdx.y;
    const int slab = blockIdx.x;          // 0..31 -> 128 rows each
    const int tid  = threadIdx.x;
    const int wave = tid >> 5;
    const int lane = tid & 31;
    const int hi   = lane >> 4;           // lane half: K=2,3 slots (zero-pad)
    const int lc   = lane & 15;           // row/col index within tile

    for (int i = tid; i < NP; i += 256) colminLDS[i] = INF_BITS;
    __syncthreads();

    const int p0 = slab * 128 + wave * 16;

    // --- A tile: rows p0..p0+15, K = (x, y, 0, 0), pre-scaled by -2 --------
    const float* xb = xs + (size_t)n * NP * 2;
    float2 xp = *(const float2*)(xb + (size_t)(p0 + lc) * 2);
    float  x2 = xp.x * xp.x + xp.y * xp.y;        // |x_p|^2, valid in lane lc
    v2f a;
    a.x = hi ? 0.0f : -2.0f * xp.x;               // lanes 0-15: K0,K1
    a.y = hi ? 0.0f : -2.0f * xp.y;               // lanes 16-31: K2,K3 = 0

    // |x|^2 for the 8 rows this lane's C fragment covers (rows r + 8*hi)
    float ax2[8];
#pragma unroll
    for (int r = 0; r < 8; ++r) ax2[r] = __shfl(x2, r + 8 * hi, 32);

    float rowacc[8];
#pragma unroll
    for (int r = 0; r < 8; ++r) rowacc[r] = 3.0e38f;

    const float* yb = ys + (size_t)n * NP * 2;
    for (int q0 = 0; q0 < NP; q0 += 16) {
        // --- B tile: cols q0..q0+15, K = (yx, yy, 0, 0) --------------------
        float2 yp = *(const float2*)(yb + (size_t)(q0 + lc) * 2);
        float  by2 = yp.x * yp.x + yp.y * yp.y;   // |y_q|^2 for this column
        v2f b;
        b.x = hi ? 0.0f : yp.x;
        b.y = hi ? 0.0f : yp.y;

        v8f c = {};
        // D = A x B (+0): 16x16 tile of -2<x_p, y_q>
        c = __builtin_amdgcn_wmma_f32_16x16x4_f32(
                /*neg_a=*/false, a, /*neg_b=*/false, b,
                /*c_mod=*/(short)0, c, /*reuse_a=*/false, /*reuse_b=*/false);

        float cmin = 3.0e38f;
#pragma unroll
        for (int r = 0; r < 8; ++r) {
            float d = c[r] + (ax2[r] + by2);      // |x|^2+|y|^2-2<x,y>
            d = fmaxf(d, EPS);                    // clip(.., EPS)
            rowacc[r] = fminf(rowacc[r], d);      // min over q (this lane's cols)
            cmin      = fminf(cmin, d);           // min over this fragment's rows
        }
        // column min (bit-pattern int min == float min for positives)
        atomicMin(&colminLDS[q0 + lc], __float_as_int(cmin));
    }

    // --- row-min across the 16 lanes of each half (xor stays inside half) --
#pragma unroll
    for (int r = 0; r < 8; ++r) {
        float v = rowacc[r];
        v = fminf(v, __shfl_xor(v, 1, 32));
        v = fminf(v, __shfl_xor(v, 2, 32));
        v = fminf(v, __shfl_xor(v, 4, 32));
        v = fminf(v, __shfl_xor(v, 8, 32));
        rowacc[r] = v;
    }
    if (lc == 0) {  // lane 0 -> rows p0..p0+7, lane 16 -> rows p0+8..p0+15
        float* dst = rowmin_g + (size_t)n * NP + p0 + hi * 8;
#pragma unroll
        for (int r = 0; r < 8; ++r) dst[r] = rowacc[r];
    }

    __syncthreads();
    int* cg = colmin_g + (size_t)n * NP;
    for (int i = tid; i < NP; i += 256) atomicMin(&cg[i], colminLDS[i]);
}

// ---------------------------------------------------------------------------
// Per-batch finalize: sqrt + means (sqrt hoisted out of the NxPxP loop via
// monotonicity), mask from sum(set2) >= 0. Fixed-order tree reduction.
// ---------------------------------------------------------------------------
__global__ __launch_bounds__(256) void cham_finalize(const float* __restrict__ rowmin,
                                                     const int*   __restrict__ colmin,
                                                     const float* __restrict__ ys,
                                                     float* __restrict__ batchcost) {
    const int n   = blockIdx.x;
    const int tid = threadIdx.x;
    float s1 = 0.0f, s2 = 0.0f, ms = 0.0f;
    for (int i = tid; i < NP; i += 256) {
        s1 += sqrtf(rowmin[(size_t)n * NP + i]);
        s2 += sqrtf(__int_as_float(colmin[(size_t)n * NP + i]));
    }
    for (int i = tid; i < NP * 2; i += 256) ms += ys[(size_t)n * NP * 2 + i];

    __shared__ float r1[256], r2[256], r3[256];
    r1[tid] = s1; r2[tid] = s2; r3[tid] = ms;
    __syncthreads();
    for (int s = 128; s > 0; s >>= 1) {
        if (tid < s) {
            r1[tid] += r1[tid + s];
            r2[tid] += r2[tid + s];
            r3[tid] += r3[tid + s];
        }
        __syncthreads();
    }
    if (tid == 0) {
        float cost = 0.5f * (r1[0] + r2[0]) * (1.0f / (float)NP);
        batchcost[n] = (r3[0] >= 0.0f) ? cost : 0.0f;
    }
}

__global__ void cham_combine(const float* __restrict__ batchcost,
                             float* __restrict__ out) {
    if (threadIdx.x == 0 && blockIdx.x == 0) {
        float s = 0.0f;
        for (int i = 0; i < NB; ++i) s += batchcost[i];
        out[0] = s * (1.0f / (float)NB);
    }
}

// ---------------------------------------------------------------------------
extern "C" void kernel_launch(void* const* d_in, const int* in_sizes, int n_in,
                              void* d_out, int out_size, void* d_ws, size_t ws_size,
                              hipStream_t stream) {
    const float* xs = (const float*)d_in[0];   // [16, 4096, 2] f32
    const float* ys = (const float*)d_in[1];   // [16, 4096, 2] f32
    float* out = (float*)d_out;                // scalar f32

    float* wsf       = (float*)d_ws;
    int*   colmin    = (int*)wsf;              // NB*NP ints   (256 KB)
    float* rowmin    = wsf + NB * NP;          // NB*NP floats (256 KB)
    float* batchcost = wsf + 2 * NB * NP;      // NB floats

    cham_init<<<(NB * NP + 255) / 256, 256, 0, stream>>>(colmin);

    dim3 grid(NP / 128, NB);                   // 32 slabs x 16 batches
    cham_main<<<grid, 256, 0, stream>>>(xs, ys, colmin, rowmin);

    cham_finalize<<<NB, 256, 0, stream>>>(rowmin, colmin, ys, batchcost);
    cham_combine<<<1, 32, 0, stream>>>(batchcost, out);
}